// CLD_86672440033829
// MI455X (gfx1250) — compile-verified
//
#include <hip/hip_runtime.h>
#include <hip/hip_bf16.h>
#include <math.h>

typedef __attribute__((ext_vector_type(2))) float v2f;
typedef __attribute__((ext_vector_type(8))) float v8f;

#define NPIX   65536
#define BATCH  64

// workspace layout (floats)
#define WS_TRACE    0
#define WS_INVNORM  1
#define RAWG_BASE   64
#define RAWG_STRIDE 8
#define PQS_BASE    (RAWG_BASE + NPIX * RAWG_STRIDE)
#define PQS_STRIDE  28
#define RSUM_BASE   (PQS_BASE + NPIX * PQS_STRIDE)

#define M_INV        4.0f
#define BETA_COEFF   8.0f          /* 4.0 * sqrt(4.0) */
#define GAMMA_COEFF  0.08f         /* 0.04 * sqrt(4.0) */
#define BG_SCALE     1.1313708498984760f /* sqrt(2*8*0.08) */
#define K_DECAY      4.5f
#define RIEMANN_MIX  0.5f

__global__ void cld_init_kernel(float* ws) {
  if (threadIdx.x == 0) ws[WS_TRACE] = 0.0f;
}

// Per-pixel covariance G = S^T S / B via V_WMMA_F32_16X16X4_F32.
// Block = 256 threads (8 waves), 32 pixels/block.
// score_x staged into LDS with GLOBAL_LOAD_ASYNC_TO_LDS_B128 (ASYNCcnt path).
__global__ void __launch_bounds__(256)
cld_cov_wmma_kernel(const float* __restrict__ score, float* __restrict__ ws) {
  __shared__ float sbuf[BATCH * 3 * 32];   // [b*3+ch][pix], 192 rows x 128B = 24 KB
  __shared__ float trace_acc;

  const int tid  = threadIdx.x;
  const int lane = tid & 31;
  const int wave = tid >> 5;
  const int p0   = blockIdx.x * 32;
  if (tid == 0) trace_acc = 0.0f;

  // Async stage: each lane moves 16B; 32 lanes cover 4 rows per instruction;
  // 8 waves -> 32 rows per sweep; 6 sweeps for 192 rows.
  {
    const int rsub = lane >> 3;             // row within 4-row group
    const int boff = (lane & 7) * 16;       // byte offset within 128B row
    // low 32 bits of a generic shared address == LDS byte offset (ISA flat rules)
    const unsigned lds_base = (unsigned)(uintptr_t)(void*)sbuf;
    const unsigned long long g_base =
        (unsigned long long)(uintptr_t)score + (unsigned long long)p0 * 4ull;
    for (int r0 = wave * 4; r0 < BATCH * 3; r0 += 32) {
      const int row = r0 + rsub;
      const unsigned lds_off = lds_base + (unsigned)(row * 128 + boff);
      const unsigned long long gaddr =
          g_base + (unsigned long long)row * (NPIX * 4ull) + (unsigned)boff;
      asm volatile("global_load_async_to_lds_b128 %0, %1, off"
                   :: "v"(lds_off), "v"(gaddr)
                   : "memory");
    }
    asm volatile("s_wait_asynccnt 0x0" ::: "memory");
  }
  __syncthreads();

  const int half = lane >> 4;       // K-halves of the 16x4 A fragment
  const int m    = lane & 15;       // M (channel) row for this lane
  const bool chOK = (m < 3);
  const int mm   = chOK ? m : 0;

  for (int pi = 0; pi < 4; ++pi) {
    const int pix = wave * 4 + pi;
    v8f acc = {};
#pragma unroll
    for (int bc = 0; bc < 16; ++bc) {
      const int b0 = bc * 4 + half * 2;
      float va = sbuf[((b0 + 0) * 3 + mm) * 32 + pix];
      float vb = sbuf[((b0 + 1) * 3 + mm) * 32 + pix];
      v2f f;
      f.x = chOK ? va : 0.0f;
      f.y = chOK ? vb : 0.0f;
      // D[i][j] += sum_k s[b_k, i] * s[b_k, j]   (A fragment == B fragment)
      acc = __builtin_amdgcn_wmma_f32_16x16x4_f32(
          false, f, false, f, (short)0, acc, false, false);
    }
    // D layout: VGPR i holds M=i, lanes 0..15 are N=0..15 -> lanes 0..2 hold G[:,lane]
    const int p = p0 + pix;
    if (lane < 3) {
      const float invB = 1.0f / 64.0f;
      float g0 = acc[0] * invB;   // G[0][lane]
      float g1 = acc[1] * invB;   // G[1][lane]
      float g2 = acc[2] * invB;   // G[2][lane]
      float* raw = ws + RAWG_BASE + (size_t)p * RAWG_STRIDE;
      if (lane == 0) { raw[0] = g0; }
      if (lane == 1) { raw[1] = g0; raw[3] = g1; }
      if (lane == 2) { raw[2] = g0; raw[4] = g1; raw[5] = g2; }
      float tr = (lane == 0) ? g0 : (lane == 1) ? g1 : g2;   // diagonal entry
      atomicAdd(&trace_acc, tr);
    }
  }
  __syncthreads();
  if (tid == 0) atomicAdd(&ws[WS_TRACE], trace_acc);
}

__global__ void cld_norm_kernel(const float* __restrict__ t, float* __restrict__ ws) {
  if (threadIdx.x == 0 && blockIdx.x == 0) {
    float t0 = t[0];
    float diag_mean = ws[WS_TRACE] / (float)NPIX / 3.0f;
    float normalization = (t0 == 1.0f) ? diag_mean * M_INV : 1.0f;
    ws[WS_INVNORM] = 1.0f / normalization;
  }
}

// Per-pixel: mix, 3x3 symmetric inverse, Cholesky, fused matrices + L row-sums.
__global__ void __launch_bounds__(256)
cld_matprep_kernel(const float* __restrict__ t, float* __restrict__ ws) {
  const int p = blockIdx.x * 256 + threadIdx.x;
  const float t0 = t[0];
  const float alpha = RIEMANN_MIX * expf(-K_DECAY * (1.0f - t0));
  const float a = alpha * ws[WS_INVNORM];
  const float cd = (1.0f - alpha) * (1.0f / M_INV);

  const float* raw = ws + RAWG_BASE + (size_t)p * RAWG_STRIDE;
  float g00 = a * raw[0] + cd, g01 = a * raw[1], g02 = a * raw[2];
  float g11 = a * raw[3] + cd, g12 = a * raw[4], g22 = a * raw[5] + cd;

  // symmetric 3x3 inverse (adjugate)
  float i00 = g11 * g22 - g12 * g12;
  float i01 = g02 * g12 - g01 * g22;
  float i02 = g01 * g12 - g02 * g11;
  float i11 = g00 * g22 - g02 * g02;
  float i12 = g01 * g02 - g00 * g12;
  float i22 = g00 * g11 - g01 * g01;
  float id  = 1.0f / (g00 * i00 + g01 * i01 + g02 * i02);
  i00 *= id; i01 *= id; i02 *= id; i11 *= id; i12 *= id; i22 *= id;

  // Cholesky lower L
  float l00 = sqrtf(g00);
  float il00 = 1.0f / l00;
  float l10 = g01 * il00;
  float l20 = g02 * il00;
  float l11 = sqrtf(g11 - l10 * l10);
  float il11 = 1.0f / l11;
  float l21 = (g12 - l20 * l10) * il11;
  float l22 = sqrtf(g22 - l20 * l20 - l21 * l21);

  float* o = ws + PQS_BASE + (size_t)p * PQS_STRIDE;
  const float B_ = BETA_COEFF, BG = BETA_COEFF * GAMMA_COEFF;
  // P = beta * L * Ginv   (drift_x = P r)
  o[0] = B_ * (l00 * i00); o[1] = B_ * (l00 * i01); o[2] = B_ * (l00 * i02);
  o[3] = B_ * (l10 * i00 + l11 * i01);
  o[4] = B_ * (l10 * i01 + l11 * i11);
  o[5] = B_ * (l10 * i02 + l11 * i12);
  o[6] = B_ * (l20 * i00 + l21 * i01 + l22 * i02);
  o[7] = B_ * (l20 * i01 + l21 * i11 + l22 * i12);
  o[8] = B_ * (l20 * i02 + l21 * i12 + l22 * i22);
  // Q = beta * L (lower-tri, dense stored)
  o[9]  = B_ * l00; o[10] = 0.0f;      o[11] = 0.0f;
  o[12] = B_ * l10; o[13] = B_ * l11;  o[14] = 0.0f;
  o[15] = B_ * l20; o[16] = B_ * l21;  o[17] = B_ * l22;
  // S = beta*Gamma * G * Ginv
  o[18] = BG * (g00 * i00 + g01 * i01 + g02 * i02);
  o[19] = BG * (g00 * i01 + g01 * i11 + g02 * i12);
  o[20] = BG * (g00 * i02 + g01 * i12 + g02 * i22);
  o[21] = BG * (g01 * i00 + g11 * i01 + g12 * i02);
  o[22] = BG * (g01 * i01 + g11 * i11 + g12 * i12);
  o[23] = BG * (g01 * i02 + g11 * i12 + g12 * i22);
  o[24] = BG * (g02 * i00 + g12 * i01 + g22 * i02);
  o[25] = BG * (g02 * i01 + g12 * i11 + g22 * i12);
  o[26] = BG * (g02 * i02 + g12 * i12 + g22 * i22);
  // L row-sums (SoA, coalesced for the streaming pass)
  ws[RSUM_BASE + 0 * NPIX + p] = l00;
  ws[RSUM_BASE + 1 * NPIX + p] = l10 + l11;
  ws[RSUM_BASE + 2 * NPIX + p] = l20 + l21 + l22;
}

// Streaming pass: thread per (b, p); reads u (coalesced), matrices from L2,
// writes all 6 drift + 6 diffusion channels.
__global__ void __launch_bounds__(256)
cld_apply_kernel(const float* __restrict__ u, const float* __restrict__ ws,
                 float* __restrict__ out) {
  const long gid = (long)blockIdx.x * 256 + threadIdx.x;
  const int p = (int)(gid & (NPIX - 1));
  const int b = (int)(gid >> 16);

  const float* m  = ws + PQS_BASE + (size_t)p * PQS_STRIDE;
  const float* ub = u + (size_t)b * 6 * NPIX + p;
  const float x0 = ub[0],        x1 = ub[NPIX],     x2 = ub[2 * NPIX];
  const float r0 = ub[3 * NPIX], r1 = ub[4 * NPIX], r2 = ub[5 * NPIX];

  const float dx0 = m[0] * r0 + m[1] * r1 + m[2] * r2;
  const float dx1 = m[3] * r0 + m[4] * r1 + m[5] * r2;
  const float dx2 = m[6] * r0 + m[7] * r1 + m[8] * r2;
  const float dr0 = -(m[9]  * x0 + m[18] * r0 + m[19] * r1 + m[20] * r2);
  const float dr1 = -(m[12] * x0 + m[13] * x1 + m[21] * r0 + m[22] * r1 + m[23] * r2);
  const float dr2 = -(m[15] * x0 + m[16] * x1 + m[17] * x2 +
                      m[24] * r0 + m[25] * r1 + m[26] * r2);

  float* ob = out + (size_t)b * 6 * NPIX + p;
  ob[0] = dx0; ob[NPIX] = dx1; ob[2 * NPIX] = dx2;
  ob[3 * NPIX] = dr0; ob[4 * NPIX] = dr1; ob[5 * NPIX] = dr2;

  float* db = out + (size_t)BATCH * 6 * NPIX + (size_t)b * 6 * NPIX + p;
  db[0] = 0.0f; db[NPIX] = 0.0f; db[2 * NPIX] = 0.0f;
  db[3 * NPIX] = BG_SCALE * ws[RSUM_BASE + 0 * NPIX + p];
  db[4 * NPIX] = BG_SCALE * ws[RSUM_BASE + 1 * NPIX + p];
  db[5 * NPIX] = BG_SCALE * ws[RSUM_BASE + 2 * NPIX + p];
}

extern "C" void kernel_launch(void* const* d_in, const int* in_sizes, int n_in,
                              void* d_out, int out_size, void* d_ws, size_t ws_size,
                              hipStream_t stream) {
  const float* u     = (const float*)d_in[0];
  const float* score = (const float*)d_in[1];
  const float* t     = (const float*)d_in[2];
  float* out = (float*)d_out;
  float* ws  = (float*)d_ws;

  cld_init_kernel<<<1, 32, 0, stream>>>(ws);
  cld_cov_wmma_kernel<<<NPIX / 32, 256, 0, stream>>>(score, ws);
  cld_norm_kernel<<<1, 32, 0, stream>>>(t, ws);
  cld_matprep_kernel<<<NPIX / 256, 256, 0, stream>>>(t, ws);
  cld_apply_kernel<<<(BATCH * NPIX) / 256, 256, 0, stream>>>(u, ws, out);
}